// PointPillarsLoss_62191126446135
// MI455X (gfx1250) — compile-verified
//
#include <hip/hip_runtime.h>
#include <stdint.h>

#define BS 256
#define CPT 4                      // cells per thread
#define TILE (BS * CPT)            // 1024 cells per block per iteration
#define NBLOCKS 256

typedef float v2f __attribute__((ext_vector_type(2)));
typedef float v8f __attribute__((ext_vector_type(8)));

// ---------- CDNA5 helpers ----------

// LDS byte offset of a generic __shared__ pointer (generic -> AS(3) -> int)
__device__ __forceinline__ uint32_t lds_off_u32(const void* p) {
  return (uint32_t)(__SIZE_TYPE__)(__attribute__((address_space(3))) const char*)p;
}

// Async global -> LDS 16B copy (per-lane), tracked by ASYNCcnt.
__device__ __forceinline__ void async_load_b128(const float* g, uint32_t lds_byte_off) {
  uint64_t ga = (uint64_t)(uintptr_t)g;
  asm volatile("global_load_async_to_lds_b128 %0, %1, off"
               :: "v"(lds_byte_off), "v"(ga) : "memory");
}

#define WAIT_ASYNC_LE_3() asm volatile("s_wait_asynccnt 3" ::: "memory")
#define WAIT_ASYNC_LE_0() asm volatile("s_wait_asynccnt 0" ::: "memory")

// Full 32-lane sum using V_WMMA_F32_16X16X4_F32:
// A = ones(16x4), B vgpr0 = v, vgpr1 = 0  =>  D[m][n] = colsum(n).
// Lanes 0..15 of D-vgpr0 hold row 0 over all 16 columns; summing those 16
// values (4 xor-shuffles, values replicated per half-wave) gives the sum of
// every B entry = sum of v over all 32 lanes.
__device__ __forceinline__ float wave_sum32(float v) {
  v2f a; a[0] = 1.0f; a[1] = 1.0f;
  v2f b; b[0] = v;    b[1] = 0.0f;
  v8f c = {0.f,0.f,0.f,0.f,0.f,0.f,0.f,0.f};
  v8f d = __builtin_amdgcn_wmma_f32_16x16x4_f32(false, a, false, b,
                                                (short)0, c, false, false);
  float s = d[0];
  s += __shfl_xor(s, 1);
  s += __shfl_xor(s, 2);
  s += __shfl_xor(s, 4);
  s += __shfl_xor(s, 8);
  return s;  // total in every lane
}

// Block (256-thread) reduction of 5 accumulators; thread 0 writes out5.
__device__ __forceinline__ void block_reduce_5(const float v[5], float* out5) {
  __shared__ float red[(BS / 32) * 5];
  const int lane = threadIdx.x & 31;
  const int wv   = threadIdx.x >> 5;
#pragma unroll
  for (int k = 0; k < 5; ++k) {
    float s = wave_sum32(v[k]);           // EXEC all-ones here by construction
    if (lane == 0) red[wv * 5 + k] = s;
  }
  __syncthreads();
  if (threadIdx.x == 0) {
#pragma unroll
    for (int k = 0; k < 5; ++k) {
      float t = 0.f;
      for (int w = 0; w < BS / 32; ++w) t += red[w * 5 + k];
      out5[k] = t;
    }
  }
  __syncthreads();
}

__device__ __forceinline__ float log_sig(float x) {
  return fminf(x, 0.0f) - log1pf(expf(-fabsf(x)));
}

__device__ __forceinline__ bool box_cell(const float* g, int N, int Hh, int Ww,
                                         int i, int* b, int* gx, int* gy, int* cid) {
  float x = g[0], y = g[1];
  *cid = (int)g[7];
  *gx = (int)floorf(x / 0.4f);
  *gy = (int)floorf((y + 50.0f) / 0.4f);
  *b = i / N;
  return (x >= 0.0f) && (x < 200.0f) && (y >= -50.0f) && (y < 50.0f) &&
         (*gx >= 0) && (*gx < Ww) && (*gy >= 0) && (*gy < Hh);
}

// ---------- kernels ----------

__global__ void pp_zero(int* a, long n) {
  long i = (long)blockIdx.x * blockDim.x + threadIdx.x;
  long stride = (long)gridDim.x * blockDim.x;
  for (; i < n; i += stride) a[i] = 0;
}

// Zero posbox only at box-center cells (the only cells atomicMax'd / read).
__global__ void pp_pos_zero(const float* __restrict__ gt, int* __restrict__ posbox,
                            int Bn, int N, int Hh, int Ww) {
  int i = blockIdx.x * blockDim.x + threadIdx.x;
  if (i >= Bn * N) return;
  int b, gx, gy, cid;
  if (!box_cell(gt + (long)i * 8, N, Hh, Ww, i, &b, &gx, &gy, &cid)) return;
  posbox[(long)b * Hh * Ww + (long)gy * Ww + gx] = 0;
}

// cellinfo bits: [2:0] class mask, [4:3] dir-bin mask, [10:8] ignore-window mask
__global__ void pp_box_scatter(const float* __restrict__ gt,
                               int* __restrict__ cellinfo, int* __restrict__ posbox,
                               float* __restrict__ boxreg,
                               int Bn, int N, int Hh, int Ww) {
  int i = blockIdx.x * blockDim.x + threadIdx.x;
  if (i >= Bn * N) return;
  const float* g = gt + (long)i * 8;
  int b, gx, gy, cid;
  if (!box_cell(g, N, Hh, Ww, i, &b, &gx, &gy, &cid)) return;
  float x = g[0], y = g[1], z = g[2], l = g[3], w = g[4], h = g[5], rot = g[6];
  long hw = (long)Hh * Ww;
  long cell = (long)b * hw + (long)gy * Ww + gx;
  int dbin = (cosf(rot) >= 0.0f) ? 0 : 1;
  atomicOr(&cellinfo[cell], (1 << cid) | (1 << (3 + dbin)));
  atomicMax(&posbox[cell], i + 1);
#pragma unroll
  for (int dy = -1; dy <= 1; ++dy)
#pragma unroll
    for (int dx = -1; dx <= 1; ++dx) {
      int wy = min(max(gy + dy, 0), Hh - 1);
      int wx = min(max(gx + dx, 0), Ww - 1);
      atomicOr(&cellinfo[(long)b * hw + (long)wy * Ww + wx], 1 << (8 + cid));
    }
  float ccx = ((float)gx + 0.5f) * 0.4f;
  float ccy = -50.0f + ((float)gy + 0.5f) * 0.4f;
  float* rv = boxreg + (long)i * 7;
  rv[0] = (x - ccx) / 0.4f;
  rv[1] = (y - ccy) / 0.4f;
  rv[2] = z;
  rv[3] = logf(fmaxf(l, 1e-3f));
  rv[4] = logf(fmaxf(w, 1e-3f));
  rv[5] = logf(fmaxf(h, 1e-3f));
  rv[6] = sinf(rot);
}

// Second scatter phase: centers end as "not ignored" (cvm True), after windows.
__global__ void pp_center_clear(const float* __restrict__ gt, int* __restrict__ cellinfo,
                                int Bn, int N, int Hh, int Ww) {
  int i = blockIdx.x * blockDim.x + threadIdx.x;
  if (i >= Bn * N) return;
  int b, gx, gy, cid;
  if (!box_cell(gt + (long)i * 8, N, Hh, Ww, i, &b, &gx, &gy, &cid)) return;
  long cell = (long)b * Hh * Ww + (long)gy * Ww + gx;
  atomicAnd(&cellinfo[cell], ~(1 << (8 + cid)));
}

// Fused loss: stream cls_pred via async b128 global->LDS double buffer, 4 cells
// per thread; reg/dir touched only at positive cells.
// Accumulators: [cls_sum, vm_sum, reg_sum, dir_sum, pos_cnt].
__global__ void pp_loss_main(const float* __restrict__ cls, const float* __restrict__ regp,
                             const float* __restrict__ dirp, const int* __restrict__ cellinfo,
                             const int* __restrict__ posbox, const float* __restrict__ boxreg,
                             float* __restrict__ partials, int HW, long total) {
  __shared__ float abuf[2][3][TILE];
  float acc[5] = {0.f, 0.f, 0.f, 0.f, 0.f};

  const long strideC = (long)gridDim.x * TILE;
  const long base0 = (long)blockIdx.x * TILE + (long)threadIdx.x * CPT;
  const int niter = (int)((total + strideC - 1) / strideC);  // uniform

  auto issue = [&](long gbase, int stage) {
    long cb = (gbase <= total - CPT) ? gbase : (total - CPT);  // 4-aligned clamp
    int b = (int)(cb / HW);
    int hw = (int)(cb % HW);
    const float* p0 = cls + (long)b * 3 * HW + hw;
#pragma unroll
    for (int c = 0; c < 3; ++c)
      async_load_b128(p0 + (long)c * HW,
                      lds_off_u32(&abuf[stage][c][threadIdx.x * CPT]));
  };

  issue(base0, 0);
  for (int it = 0; it < niter; ++it) {
    long base = base0 + (long)it * strideC;
    bool more = (it + 1 < niter);              // uniform branch
    if (more) {
      issue(base + strideC, (it + 1) & 1);
      WAIT_ASYNC_LE_3();                       // previous stage's 3 loads done
    } else {
      WAIT_ASYNC_LE_0();
    }
    float4 xc[3];
#pragma unroll
    for (int c = 0; c < 3; ++c)
      xc[c] = *(const float4*)&abuf[it & 1][c][threadIdx.x * CPT];

    // base and total are both multiples of CPT: either all 4 cells valid or none.
    if (base < total) {
      int b = (int)(base / HW);
      int hw0 = (int)(base % HW);              // group never straddles a plane
      int4 infov = *(const int4*)&cellinfo[base];
      const int infos[4] = {infov.x, infov.y, infov.z, infov.w};

#pragma unroll
      for (int u = 0; u < 4; ++u) {
        int info = infos[u];
        // focal loss, 3 channels
#pragma unroll
        for (int c = 0; c < 3; ++c) {
          float xv = (c == 0) ? ((const float*)&xc[0])[u]
                   : (c == 1) ? ((const float*)&xc[1])[u]
                              : ((const float*)&xc[2])[u];
          int t = (info >> c) & 1;
          float vm = 1.0f - (float)((info >> (8 + c)) & 1);
          float lsx = log_sig(xv);
          float bce = t ? -lsx : (xv - lsx);   // ls(-x) = ls(x) - x
          float p = 1.0f / (1.0f + expf(-xv));
          float pt = t ? p : (1.0f - p);
          float uu = 1.0f - pt;
          float fw = (t ? 0.25f : 0.75f) * uu * uu * uu;
          float lossv = fw * bce;
          if (t) lossv *= (c == 0 ? 1.0f : 5.0f);
          acc[0] += lossv * vm;
          acc[1] += vm;
        }

        if (info & 7) {  // positive cell: reg smooth-L1 + dir BCE
          long idx = base + u;
          int hw = hw0 + u;
          int pb = posbox[idx] - 1;
          const float* rv = boxreg + (long)pb * 7;
          long rbase = (long)b * 7 * HW + hw;
#pragma unroll
          for (int j = 0; j < 7; ++j) {
            float d = regp[rbase + (long)j * HW] - rv[j];
            float ad = fabsf(d);
            acc[2] += (ad < 1.0f) ? 0.5f * d * d : (ad - 0.5f);
          }
          long dbase = (long)b * 2 * HW + hw;
          int dm = (info >> 3) & 3;
#pragma unroll
          for (int k = 0; k < 2; ++k) {
            float xv = dirp[dbase + (long)k * HW];
            int t = (dm >> k) & 1;
            float lsx = log_sig(xv);
            acc[3] += t ? -lsx : (xv - lsx);
          }
          acc[4] += 1.0f;
        }
      }
    }
  }

  __shared__ float out5[5];
  block_reduce_5(acc, out5);
  if (threadIdx.x == 0) {
#pragma unroll
    for (int k = 0; k < 5; ++k) partials[(long)blockIdx.x * 5 + k] = out5[k];
  }
}

__global__ void pp_finalize(const float* __restrict__ partials, int nblk,
                            float* __restrict__ out) {
  float acc[5] = {0.f, 0.f, 0.f, 0.f, 0.f};
  for (int i = threadIdx.x; i < nblk; i += blockDim.x)
#pragma unroll
    for (int k = 0; k < 5; ++k) acc[k] += partials[(long)i * 5 + k];

  __shared__ float out5[5];
  block_reduce_5(acc, out5);
  if (threadIdx.x == 0) {
    float cls_sum = out5[0], vm = out5[1], reg_sum = out5[2];
    float dir_sum = out5[3], pos = out5[4];
    float cls_loss = cls_sum / (vm + 1e-6f);
    float rcnt = pos * 7.0f;
    float reg_loss = (rcnt > 0.0f) ? reg_sum / fmaxf(rcnt, 1.0f) : 0.0f;
    float dcnt = pos * 2.0f;
    float dir_loss = (dcnt > 0.0f) ? dir_sum / fmaxf(dcnt, 1.0f) : 0.0f;
    out[0] = 1.0f * cls_loss + 2.0f * reg_loss + 0.2f * dir_loss;
    out[1] = cls_loss;
    out[2] = reg_loss;
    out[3] = dir_loss;
  }
}

// ---------- host launcher ----------

extern "C" void kernel_launch(void* const* d_in, const int* in_sizes, int n_in,
                              void* d_out, int out_size, void* d_ws, size_t ws_size,
                              hipStream_t stream) {
  const float* cls  = (const float*)d_in[0];
  const float* regp = (const float*)d_in[1];
  const float* dirp = (const float*)d_in[2];
  const float* gt   = (const float*)d_in[3];

  const int N = 64;
  int Bn = in_sizes[3] / (N * 8);
  if (Bn <= 0) Bn = 8;
  long HW = (long)in_sizes[0] / (3 * Bn);   // 125000
  int Ww = 500;
  int Hh = (int)(HW / Ww);                  // 250
  long total = (long)Bn * HW;               // 1,000,000 cells

  char* ws = (char*)d_ws;
  int*   cellinfo = (int*)ws;                                   // total ints
  int*   posbox   = (int*)(ws + 4 * total);                     // total ints
  float* boxreg   = (float*)(ws + 8 * total);                   // Bn*N*7 floats
  float* partials = (float*)(ws + 8 * total + 4L * Bn * N * 7); // NBLOCKS*5

  pp_zero<<<1024, BS, 0, stream>>>(cellinfo, total);
  int nbb = (Bn * N + BS - 1) / BS;
  pp_pos_zero<<<nbb, BS, 0, stream>>>(gt, posbox, Bn, N, Hh, Ww);
  pp_box_scatter<<<nbb, BS, 0, stream>>>(gt, cellinfo, posbox, boxreg, Bn, N, Hh, Ww);
  pp_center_clear<<<nbb, BS, 0, stream>>>(gt, cellinfo, Bn, N, Hh, Ww);
  pp_loss_main<<<NBLOCKS, BS, 0, stream>>>(cls, regp, dirp, cellinfo, posbox, boxreg,
                                           partials, (int)HW, total);
  pp_finalize<<<1, BS, 0, stream>>>(partials, NBLOCKS, (float*)d_out);
}